// Net_66554813219287
// MI455X (gfx1250) — compile-verified
//
#include <hip/hip_runtime.h>

// ---------------------------------------------------------------------------
// Problem constants (match the JAX reference)
// ---------------------------------------------------------------------------
#define CIN   32          // feature channels per image
#define D_IN  48          // MAXDISP/4
#define H_IN  64
#define W_IN  128
#define MAXD  192
#define H_OUT 256         // 4*H_IN
#define W_OUT 512         // 4*W_IN

static constexpr long long COST_ELEMS = (long long)(2 * CIN) * D_IN * H_IN * W_IN; // 25,165,824

typedef __attribute__((ext_vector_type(2))) float v2f;
typedef __attribute__((ext_vector_type(8))) float v8f;

// ---------------------------------------------------------------------------
// Kernel 1: cost volume. 100.7 MB of stores, ~2 MB of (cached) loads.
// One float4 store per thread; out linear index == t*4 by construction.
//   out[c,d,h,w] = (c<32) ? (w>=d ? left[c,h,w]      : 0)
//                         : (w>=d ? right[c-32,h,w-d]: 0)
// ---------------------------------------------------------------------------
__global__ __launch_bounds__(256) void cost_volume_kernel(
    const float* __restrict__ left, const float* __restrict__ right,
    float* __restrict__ out)
{
    int t   = blockIdx.x * blockDim.x + threadIdx.x;   // float4 index
    int w4  = t & 31;          // 32 float4 per row of 128
    int rst = t >> 5;
    int h   = rst & 63;        rst >>= 6;
    int d   = rst % D_IN;
    int c   = rst / D_IN;      // 0..63
    int w   = w4 * 4;

    float4 v;
    if (c < CIN) {
        // left half: aligned vector load, mask w<d to zero
        const float4 L = *reinterpret_cast<const float4*>(left + ((size_t)(c * H_IN + h) * W_IN + w));
        v.x = (w + 0 >= d) ? L.x : 0.0f;
        v.y = (w + 1 >= d) ? L.y : 0.0f;
        v.z = (w + 2 >= d) ? L.z : 0.0f;
        v.w = (w + 3 >= d) ? L.w : 0.0f;
    } else {
        // right half: shifted (unaligned) gather from a 1 MB tensor -> L1/L2 hits
        const float* r = right + (size_t)((c - CIN) * H_IN + h) * W_IN;
        v.x = (w + 0 >= d) ? r[w + 0 - d] : 0.0f;
        v.y = (w + 1 >= d) ? r[w + 1 - d] : 0.0f;
        v.z = (w + 2 >= d) ? r[w + 2 - d] : 0.0f;
        v.w = (w + 3 >= d) ? r[w + 3 - d] : 0.0f;
    }
    *reinterpret_cast<float4*>(out + (size_t)t * 4) = v;
}

// ---------------------------------------------------------------------------
// Kernel 2: fused trilinear-x4 upsample + softmax + soft-argmax.
// Per wave: 16 output pixels. The disparity-axis upsample is the linear map
//   D(192x16) = U(192x48) x Col(48x16)
// run on V_WMMA_F32_16X16X4_F32 (full fp32). Key layout facts (ISA 7.12.2):
//   A (16x4 f32): lanes 0-15 hold K={kb,kb+1}, lanes 16-31 hold K={kb+2,kb+3},
//                 M = lane&15.  B (4x16 f32): same K split, N = lane&15.
// => lane half hi only ever feeds col[4j+2*hi] / col[4j+2*hi+1]. Compute ONLY
// those 24 values per lane, indexed by the unrolled constant j, so everything
// stays in VGPRs (no dynamic indexing -> no scratch spills).
// U rows have 2 non-zeros, so all-zero A tiles are skipped at compile time.
// ---------------------------------------------------------------------------
constexpr int kcol_min(int mt) { return (mt * 16 * 47) / 191 - 1; }          // conservative
constexpr int kcol_max(int mt) { return ((mt * 16 + 15) * 47) / 191 + 2; }   // conservative

__global__ __launch_bounds__(256) void disp_regression_kernel(
    const float* __restrict__ hg, float* __restrict__ out)
{
    const int gtid = blockIdx.x * blockDim.x + threadIdx.x;
    const int lane = gtid & 31;
    const int wave = gtid >> 5;
    const int n    = lane & 15;   // pixel within the 16-pixel group (matrix column)
    const int hi   = lane >> 4;   // which half of the wave (K pairs / M+8 rows)

    const int pix   = wave * 16 + n;        // 0 .. 131071
    const int h_out = pix >> 9;             // /512 (512 % 16 == 0 -> group shares h)
    const int w_out = pix & 511;

    // align_corners bilinear source coords + weights (matches reference)
    float ph = (float)h_out * (63.0f / 255.0f);
    float phf = floorf(ph);
    int   h0 = (int)phf;           if (h0 > H_IN - 1) h0 = H_IN - 1;
    int   h1 = min(h0 + 1, H_IN - 1);
    float fh = ph - (float)h0;

    float pw = (float)w_out * (127.0f / 511.0f);
    float pwf = floorf(pw);
    int   w0 = (int)pwf;           if (w0 > W_IN - 1) w0 = W_IN - 1;
    int   w1 = min(w0 + 1, W_IN - 1);
    float fw = pw - (float)w0;

    const float c00 = (1.0f - fh) * (1.0f - fw);
    const float c01 = (1.0f - fh) * fw;
    const float c10 = fh * (1.0f - fw);
    const float c11 = fh * fw;

    const int o00 = h0 * W_IN + w0, o01 = h0 * W_IN + w1;
    const int o10 = h1 * W_IN + w0, o11 = h1 * W_IN + w1;

    // B-operand columns this lane actually feeds: col[4j+2*hi], col[4j+2*hi+1].
    // Runtime 'hi' lives only in the load ADDRESS; bx/by use constant indices.
    const size_t HW = (size_t)H_IN * W_IN;
    float bx[12], by[12];
#pragma unroll
    for (int j = 0; j < 12; ++j) {
        const float* p0 = hg + (size_t)(j * 4 + 2 * hi) * HW;
        const float* p1 = p0 + HW;
        bx[j] = c00 * p0[o00] + c01 * p0[o01] + c10 * p0[o10] + c11 * p0[o11];
        by[j] = c00 * p1[o00] + c01 * p1[o01] + c10 * p1[o10] + c11 * p1[o11];
    }

    // Per-M-tile disparity interpolation weights (constant per lane, hoisted)
    int   d0v[12];
    float fdv[12];
#pragma unroll
    for (int mt = 0; mt < 12; ++mt) {
        float pd  = (float)(mt * 16 + n) * (47.0f / 191.0f);
        float pdf = floorf(pd);
        d0v[mt] = (int)pdf;
        fdv[mt] = pd - pdf;
    }

    // 12 accumulator tiles: up[mt*16 .. mt*16+15][pixel 0..15]
    v8f acc[12];
    const v8f vzero = {0.f, 0.f, 0.f, 0.f, 0.f, 0.f, 0.f, 0.f};
#pragma unroll
    for (int mt = 0; mt < 12; ++mt) acc[mt] = vzero;

#pragma unroll
    for (int j = 0; j < 12; ++j) {
        const int kb = j * 4;
        v2f B;
        B.x = bx[j];
        B.y = by[j];
        const int c0 = kb + 2 * hi;   // K column of A.x for this lane
        const int c1 = c0 + 1;        // K column of A.y
#pragma unroll
        for (int mt = 0; mt < 12; ++mt) {
            if (kb + 3 < kcol_min(mt) || kb > kcol_max(mt)) continue; // all-zero A tile
            const int   d0 = d0v[mt];
            const float fd = fdv[mt];
            v2f A;
            A.x = (c0 == d0) ? (1.0f - fd) : ((c0 == d0 + 1) ? fd : 0.0f);
            A.y = (c1 == d0) ? (1.0f - fd) : ((c1 == d0 + 1) ? fd : 0.0f);
            acc[mt] = __builtin_amdgcn_wmma_f32_16x16x4_f32(
                false, A, false, B, (short)0, acc[mt], false, false);
        }
    }

    // D layout: tile mt, vgpr r -> dout = mt*16 + (hi?8:0) + r, pixel = lane&15.
    // Lane n & lane n+16 jointly hold all 192 disparities of pixel n.
    float M = -3.402823466e38f;
#pragma unroll
    for (int mt = 0; mt < 12; ++mt)
#pragma unroll
        for (int r = 0; r < 8; ++r) M = fmaxf(M, acc[mt][r]);
    M = fmaxf(M, __shfl_xor(M, 16, 32));

    const int dbase = hi ? 8 : 0;
    float se = 0.0f, sw = 0.0f;
#pragma unroll
    for (int mt = 0; mt < 12; ++mt) {
#pragma unroll
        for (int r = 0; r < 8; ++r) {
            float e = __expf(acc[mt][r] - M);
            se += e;
            sw += e * (float)(mt * 16 + dbase + r);
        }
    }
    se += __shfl_xor(se, 16, 32);
    sw += __shfl_xor(sw, 16, 32);

    if (hi == 0) out[pix] = sw / se;
}

// ---------------------------------------------------------------------------
// Launch: inputs are [left(1,32,64,128), right(1,32,64,128), hourglass(1,1,48,64,128)],
// output is cost(25,165,824) ++ disp(131,072), all f32.
// ---------------------------------------------------------------------------
extern "C" void kernel_launch(void* const* d_in, const int* in_sizes, int n_in,
                              void* d_out, int out_size, void* d_ws, size_t ws_size,
                              hipStream_t stream) {
    (void)in_sizes; (void)n_in; (void)out_size; (void)d_ws; (void)ws_size;
    const float* left  = (const float*)d_in[0];
    const float* right = (const float*)d_in[1];
    const float* hg    = (const float*)d_in[2];
    float* out = (float*)d_out;

    // Cost volume: 6,291,456 float4 stores
    const int nvec = (int)(COST_ELEMS / 4);
    cost_volume_kernel<<<nvec / 256, 256, 0, stream>>>(left, right, out);

    // Disparity regression: 131072 pixels, 2 lanes/pixel -> 262144 threads
    float* disp = out + (size_t)COST_ELEMS;
    disp_regression_kernel<<<(H_OUT * W_OUT * 2) / 256, 256, 0, stream>>>(hg, disp);
}